// Model_26637387170046
// MI455X (gfx1250) — compile-verified
//
#include <hip/hip_runtime.h>

// Problem constants (match reference)
constexpr int kB      = 2;
constexpr int kSQ     = 512;
constexpr int kNH     = 16;    // heads
constexpr int kDH     = 128;   // head dim
constexpr int kBLOCK  = 128;   // paged-cache block size
constexpr int kMAXBLK = 32;
constexpr int kSKV    = 4096;  // logical KV length
constexpr int kSPARSE = 2048;  // top-k

constexpr int kQPB    = 8;               // queries per block (one per wave)
constexpr int kKSPLIT = 2;               // k-range splits for grid spread
constexpr int kTILES  = kSKV / 16;       // 256 K-tiles of 16 kv positions
constexpr int kTPB    = kTILES / kKSPLIT;// tiles per block = 128
constexpr int kROWSTR = 132;             // padded LDS row stride (floats) -> no bank conflicts

typedef float v2f __attribute__((ext_vector_type(2)));
typedef float v8f __attribute__((ext_vector_type(8)));
typedef int   v4i __attribute__((vector_size(16)));   // matches builtin param type

typedef __attribute__((address_space(1))) v4i* as1_v4i_ptr;
typedef __attribute__((address_space(3))) v4i* as3_v4i_ptr;

// ---------------------------------------------------------------------------
// CDNA5 async global->LDS staging (ASYNCcnt path), with a safe fallback.
// ---------------------------------------------------------------------------
#if defined(__gfx1250__) && \
    __has_builtin(__builtin_amdgcn_global_load_async_to_lds_b128) && \
    __has_builtin(__builtin_amdgcn_s_wait_asynccnt)
#define USE_ASYNC_LDS 1
#else
#define USE_ASYNC_LDS 0
#endif

__device__ __forceinline__ void copy16_to_lds(float* l, const float* g) {
#if USE_ASYNC_LDS
    // Generic shared addr low 32 bits == LDS offset.
    __builtin_amdgcn_global_load_async_to_lds_b128(
        (as1_v4i_ptr)(unsigned long long)(const void*)g,
        (as3_v4i_ptr)(unsigned)(unsigned long long)(void*)l,
        /*offset=*/0, /*cpol=*/0);
#else
    *(float4*)l = *(const float4*)g;
#endif
}

__device__ __forceinline__ void wait_copies() {
#if USE_ASYNC_LDS
    __builtin_amdgcn_s_wait_asynccnt(0);
#endif
}

// ---------------------------------------------------------------------------
// Kernel 1: scores[b,q,k] = sum_n w[b,q,n] * relu( q[b,q,n,:] . kv[b,k,:] )
// Block = 8 waves = 8 query tokens sharing double-buffered KV tiles in LDS.
// WMMA tile: M = 16 heads, N = 16 kv positions, K = D via 32 chained
// V_WMMA_F32_16X16X4_F32 (pure fp32 -- matches the f32 reference).
// ---------------------------------------------------------------------------
__global__ __launch_bounds__(32 * kQPB)
void idx_scores_kernel(const float* __restrict__ query,    // [B,SQ,NH,DH]
                       const float* __restrict__ kcache,   // [NBLK,BLOCK,1,DH]
                       const float* __restrict__ weights,  // [B,SQ,NH]
                       const int*   __restrict__ btab,     // [B,MAXBLK]
                       float*       __restrict__ scores)   // [B*SQ, SKV]
{
    __shared__ float tile[2][16 * kROWSTR];      // 2 x 8448 B, 16-byte aligned rows

    const int b     = blockIdx.x / (kSQ / kQPB);
    const int qbase = (blockIdx.x % (kSQ / kQPB)) * kQPB;
    const int tbase = blockIdx.y * kTPB;

    const int tid  = threadIdx.x;
    const int lane = tid & 31;
    const int wave = tid >> 5;           // 0..7 -> this wave's query token
    const int m    = lane & 15;          // A-row (head) / B-col (kv pos in tile)
    const int hi   = lane >> 4;          // half-wave selector (K pairs)

    const int row = b * kSQ + qbase + wave;      // (b,q) row id

    // ---- A operand: this wave's query tile (16 heads x 128 dims), WMMA layout.
    const float* qrow = query + ((size_t)row * kNH + m) * kDH + 2 * hi;
    v2f A[32];
#pragma unroll
    for (int kk = 0; kk < 32; ++kk) {
        A[kk].x = qrow[4 * kk + 0];
        A[kk].y = qrow[4 * kk + 1];
    }

    // Weights for the 8 head-rows this lane accumulates (C VGPR v -> head v+8*hi).
    const float* wrow = weights + (size_t)row * kNH + 8 * hi;
    float w8[8];
#pragma unroll
    for (int v = 0; v < 8; ++v) w8[v] = wrow[v];

    float* srow = scores + (size_t)row * kSKV;

    // Cooperative tile copy: wave w stages rows 2w and 2w+1 (16 B per lane,
    // 32 lanes x 16 B = one full 512 B KV row per instruction).
    auto copy_tile = [&](int t, int bufsel) {
        const int k0  = 16 * t;
        const int blk = btab[b * kMAXBLK + (k0 >> 7)];  // tiles never cross pages
        const float* gbase = kcache + ((size_t)blk * kBLOCK + (k0 & 127)) * kDH;
#pragma unroll
        for (int j = 0; j < 2; ++j) {
            const int r = 2 * wave + j;
            copy16_to_lds(&tile[bufsel][r * kROWSTR + lane * 4],
                          gbase + (size_t)r * kDH + lane * 4);
        }
    };

    // Prologue: fill both buffers.
    copy_tile(tbase + 0, 0);
    copy_tile(tbase + 1, 1);
    wait_copies();
    __syncthreads();

    for (int i = 0; i < kTPB; ++i) {
        const int t = tbase + i;
        const float* buf = tile[i & 1];

        v8f c = {};
#pragma unroll
        for (int kk = 0; kk < 32; ++kk) {
            // B operand 4x16 (K x N): col = kv pos m, row stride 132 floats ->
            // lanes hit all 64 LDS banks exactly once per ds_load_b64.
            v2f bb = *(const v2f*)&buf[m * kROWSTR + 4 * kk + 2 * hi];
            c = __builtin_amdgcn_wmma_f32_16x16x4_f32(
                    /*neg_a=*/false, A[kk], /*neg_b=*/false, bb,
                    /*c_mod=*/(short)0, c, /*reuse_a=*/false, /*reuse_b=*/false);
        }

        // Head reduction: relu * weight over 8 accumulator VGPRs, then fold
        // the two lane halves (heads 0..7 vs 8..15).
        float p = 0.f;
#pragma unroll
        for (int v = 0; v < 8; ++v) {
            float x = c[v];
            x = x > 0.f ? x : 0.f;
            p += x * w8[v];
        }
        p += __shfl_xor(p, 16, 32);
        if (lane < 16) srow[16 * t + m] = p;

        __syncthreads();                        // all waves done reading buf
        if (i + 2 < kTPB) copy_tile(t + 2, i & 1);
        wait_copies();
        __syncthreads();                        // refilled buffer visible
    }
}

// ---------------------------------------------------------------------------
// Kernel 2: exact top-2048-of-4096, sorted descending (ties -> lower index),
// via in-LDS bitonic sort of packed u64 keys: (~orderedFloat(val) << 32)|idx.
// ---------------------------------------------------------------------------
__global__ __launch_bounds__(512)
void topk_kernel(const float* __restrict__ scores,   // [B*SQ, SKV]
                 const int*   __restrict__ seqlens,  // [B]
                 float*       __restrict__ out_vals, // [B*SQ, SPARSE]
                 int*         __restrict__ out_idx)  // [B*SQ, SPARSE]
{
    __shared__ unsigned long long keys[kSKV];        // 32 KB
    const int row  = blockIdx.x;
    const int b    = row / kSQ;
    const int q    = row % kSQ;
    const int L    = seqlens[b];
    const int qpos = L - kSQ + q;                    // causal boundary
    const int tid  = threadIdx.x;
    const float* srow = scores + (size_t)row * kSKV;

#pragma unroll
    for (int i = 0; i < kSKV / 512; ++i) {
        const int k = tid + 512 * i;
        const float v = srow[k];
        unsigned vb = __float_as_uint(v);
        unsigned ov = vb ^ ((vb & 0x80000000u) ? 0xFFFFFFFFu : 0x80000000u);
        const bool valid = (k <= qpos) && (k < L);
        if (!valid) ov = 0u;                         // sorts to the very end
        keys[k] = ((unsigned long long)(~ov) << 32) | (unsigned)k;
    }
    __syncthreads();

    for (int k2 = 2; k2 <= kSKV; k2 <<= 1) {
        for (int j = k2 >> 1; j > 0; j >>= 1) {
            for (int e = tid; e < kSKV; e += 512) {
                const int x = e ^ j;
                if (x > e) {
                    const unsigned long long a  = keys[e];
                    const unsigned long long bk = keys[x];
                    const bool up = ((e & k2) == 0);
                    if ((a > bk) == up) { keys[e] = bk; keys[x] = a; }
                }
            }
            __syncthreads();
        }
    }

#pragma unroll
    for (int i = 0; i < kSPARSE / 512; ++i) {
        const int r = tid + 512 * i;
        const unsigned long long key = keys[r];
        const unsigned idx = (unsigned)(key & 0xFFFFFFFFu);
        const unsigned ov  = ~(unsigned)(key >> 32);
        const unsigned bits = (ov & 0x80000000u) ? (ov ^ 0x80000000u) : ~ov;
        out_vals[(size_t)row * kSPARSE + r] = __uint_as_float(bits);
        out_idx [(size_t)row * kSPARSE + r] = (int)idx;
    }
}

// ---------------------------------------------------------------------------
extern "C" void kernel_launch(void* const* d_in, const int* in_sizes, int n_in,
                              void* d_out, int out_size, void* d_ws, size_t ws_size,
                              hipStream_t stream) {
    const float* query   = (const float*)d_in[0];  // [B,SQ,N,D]
    const float* kcache  = (const float*)d_in[1];  // [NUM_BLOCKS,BLOCK,1,D]
    const float* weights = (const float*)d_in[2];  // [B,SQ,N]
    const int*   seqlens = (const int*)d_in[3];    // [B]
    const int*   btab    = (const int*)d_in[4];    // [B,MAX_BLK]

    float* scores = (float*)d_ws;                  // B*SQ*SKV floats = 16 MB

    float* out_vals = (float*)d_out;                                // first output
    int*   out_idx  = (int*)d_out + (size_t)kB * kSQ * kSPARSE;     // second output

    dim3 grid1(kB * (kSQ / kQPB), kKSPLIT);        // 128 x 2 blocks, 256 thr
    idx_scores_kernel<<<grid1, 32 * kQPB, 0, stream>>>(query, kcache, weights,
                                                       btab, scores);
    topk_kernel<<<kB * kSQ, 512, 0, stream>>>(scores, seqlens, out_vals, out_idx);
}